// BackMapLayer_57019985821997
// MI455X (gfx1250) — compile-verified
//
#include <hip/hip_runtime.h>
#include <math.h>

// Problem constants from the reference: B=256, N=4096.
#define BSZ   256
#define NPTS  4096
#define NDIST (NPTS - 1)   // 4095
#define NANG  (NPTS - 2)   // 4094
#define NDIH  (NPTS - 3)   // 4093
#define SPLIT (NPTS / 2)   // 2048
#define PI_F  3.14159265358979323846f

typedef __attribute__((ext_vector_type(2))) float v2f;
typedef __attribute__((ext_vector_type(8))) float v8f;

#define LDS_AS __attribute__((address_space(3)))
#define GLB_AS __attribute__((address_space(1)))

static __device__ __forceinline__ float wbcast(float v, int lane) {
  return __shfl(v, lane, 32);
}
static __device__ __forceinline__ float wup(float v, int off) {
  return __shfl_up(v, off, 32);
}

// ---- gfx1250 async global->LDS copy (ASYNCcnt-tracked), with fallbacks -----
static __device__ __forceinline__ void async_g2l_b32(const float* g, float* l) {
#if __has_builtin(__builtin_amdgcn_global_load_async_to_lds_b32)
  __builtin_amdgcn_global_load_async_to_lds_b32(
      (GLB_AS int*)(float*)g, (LDS_AS int*)l, 0, 0);
#else
  unsigned loff = (unsigned)(uintptr_t)(LDS_AS float*)l;
  asm volatile("global_load_async_to_lds_b32 %0, %1, off"
               :: "v"(loff), "v"(g) : "memory");
#endif
}
static __device__ __forceinline__ void wait_async0() {
#if __has_builtin(__builtin_amdgcn_s_wait_asynccnt)
  __builtin_amdgcn_s_wait_asynccnt(0);
#else
  asm volatile("s_wait_asynccnt 0x0" ::: "memory");
#endif
}
static __device__ __forceinline__ void wait_async1() {
#if __has_builtin(__builtin_amdgcn_s_wait_asynccnt)
  __builtin_amdgcn_s_wait_asynccnt(1);
#else
  asm volatile("s_wait_asynccnt 0x1" ::: "memory");
#endif
}

// ---------------- Kernel 1: mean bond lengths over the batch ----------------
__global__ void k_mean_lengths(const float* __restrict__ dist,
                               float* __restrict__ lengths) {
  int j = blockIdx.x * blockDim.x + threadIdx.x;
  if (j >= NDIST) return;
  float s = 0.0f;
  for (int b = 0; b < BSZ; ++b) s += dist[b * NDIST + j];
  lengths[j] = s * (1.0f / (float)BSZ);
}

// ---------------- Kernel 2: planar zig-zag chain (one wave per row) ---------
__global__ void k_chain(const float* __restrict__ angles,
                        const float* __restrict__ lengths,
                        float* __restrict__ cx, float* __restrict__ cy) {
  const int lane = threadIdx.x & 31;
  const int wave = threadIdx.x >> 5;
  const int row  = blockIdx.x * (blockDim.x >> 5) + wave;
  if (row >= BSZ) return;

  const float* __restrict__ ar = angles + (size_t)row * NANG;
  float* __restrict__ rx = cx + (size_t)row * NPTS;
  float* __restrict__ ry = cy + (size_t)row * NPTS;

  if (lane == 0) { rx[0] = 0.0f; ry[0] = 0.0f; }

  float carryS = 0.0f, carryX = 0.0f, carryY = 0.0f;

  for (int c = 0; c < NPTS / 32; ++c) {           // 128 chunks
    const int j = c * 32 + lane;
    if (j + 32 < NANG) __builtin_prefetch(&ar[j + 32], 0, 0);

    float w = 0.0f;
    if (j < NANG) {
      const float alt = (j & 1) ? -1.0f : 1.0f;
      w = alt * (PI_F - ar[j]);
    }
    #pragma unroll
    for (int off = 1; off < 32; off <<= 1) {
      float t = wup(w, off);
      if (lane >= off) w += t;
    }
    const float S = carryS + w;
    float Sprev = wup(S, 1);
    if (lane == 0) Sprev = carryS;

    float p = 0.0f;
    if (j > 0) {
      const float altm1 = ((j - 1) & 1) ? -1.0f : 1.0f;
      p = altm1 * Sprev;
    }
    float sp, cp;
    sincosf(p, &sp, &cp);
    const float len = (j < NDIST) ? lengths[j] : 0.0f;
    const float seg = (j & 1) ? -1.0f : 1.0f;
    float dx = len * cp;
    float dy = len * sp * seg;

    #pragma unroll
    for (int off = 1; off < 32; off <<= 1) {
      float tx = wup(dx, off);
      float ty = wup(dy, off);
      if (lane >= off) { dx += tx; dy += ty; }
    }
    const float X = carryX + dx;
    const float Y = carryY + dy;
    if (j < NDIST) { rx[j + 1] = X; ry[j + 1] = Y; }

    carryS = wbcast(S, 31);
    carryX = wbcast(X, 31);
    carryY = wbcast(Y, 31);
  }
}

// ---------------- Kernel 3: torsion application (one wave per row-half) -----
struct Mat3 { float m[3][3]; };

static __device__ __forceinline__ Mat3 matmul3(const Mat3& A, const Mat3& B) {
  Mat3 C;
  #pragma unroll
  for (int i = 0; i < 3; ++i)
    #pragma unroll
    for (int j = 0; j < 3; ++j)
      C.m[i][j] = fmaf(A.m[i][0], B.m[0][j],
                  fmaf(A.m[i][1], B.m[1][j],
                       A.m[i][2] * B.m[2][j]));
  return C;
}
static __device__ __forceinline__ Mat3 shflup3(const Mat3& A, int off) {
  Mat3 R;
  #pragma unroll
  for (int i = 0; i < 3; ++i)
    #pragma unroll
    for (int j = 0; j < 3; ++j)
      R.m[i][j] = wup(A.m[i][j], off);
  return R;
}
static __device__ __forceinline__ Mat3 bcast3(const Mat3& A, int lane) {
  Mat3 R;
  #pragma unroll
  for (int i = 0; i < 3; ++i)
    #pragma unroll
    for (int j = 0; j < 3; ++j)
      R.m[i][j] = wbcast(A.m[i][j], lane);
  return R;
}
static __device__ __forceinline__ Mat3 ident3() {
  Mat3 I;
  #pragma unroll
  for (int i = 0; i < 3; ++i)
    #pragma unroll
    for (int j = 0; j < 3; ++j)
      I.m[i][j] = (i == j) ? 1.0f : 0.0f;
  return I;
}

// P_new = P @ L31 via v_wmma_f32_16x16x4_f32 (3x3 blocks embedded top-left).
// A(m,k): lane = m + 16*(k>=2), vgpr = k&1   (ISA 7.12.2, 32-bit A 16x4)
// B(k,n): lane = n + 16*(k>=2), vgpr = k&1   (row-striped, half-split pattern)
// D(i,j): vgpr = i, lane = j                 (ISA 7.12.2, 32-bit C/D 16x16)
static __device__ __forceinline__ Mat3 carry_update(const Mat3& P,
                                                    const Mat3& L31,
                                                    int lane) {
#if __has_builtin(__builtin_amdgcn_wmma_f32_16x16x4_f32)
  const int mrow = lane & 15;
  const int kh   = lane >> 4;
  v2f a; a.x = 0.0f; a.y = 0.0f;
  v2f b; b.x = 0.0f; b.y = 0.0f;
  if (mrow < 3) {
    a.x = kh ? P.m[mrow][2] : P.m[mrow][0];
    a.y = kh ? 0.0f         : P.m[mrow][1];
    b.x = kh ? L31.m[2][mrow] : L31.m[0][mrow];
    b.y = kh ? 0.0f           : L31.m[1][mrow];
  }
  v8f cacc = {};
  v8f d = __builtin_amdgcn_wmma_f32_16x16x4_f32(
      false, a, false, b, (short)0, cacc, false, false);
  Mat3 Pn;
  #pragma unroll
  for (int i = 0; i < 3; ++i)
    #pragma unroll
    for (int j = 0; j < 3; ++j)
      Pn.m[i][j] = wbcast(d[i], j);
  return Pn;
#else
  (void)lane;
  return matmul3(P, L31);
#endif
}

__global__ void k_torsion(const float* __restrict__ dih,
                          const float* __restrict__ cx,
                          const float* __restrict__ cy,
                          float* __restrict__ out) {
  const int lane = threadIdx.x & 31;
  const int wave = threadIdx.x >> 5;
  const int wid  = blockIdx.x * (blockDim.x >> 5) + wave;
  const int row  = wid >> 1;
  const int side = wid & 1;              // 0 = right half, 1 = left half
  if (row >= BSZ) return;

  // triple-buffered per-wave staging of dihedrals via async global->LDS
  __shared__ float dstage[8][3][32];

  const float* __restrict__ rx = cx + (size_t)row * NPTS;
  const float* __restrict__ ry = cy + (size_t)row * NPTS;
  const float* __restrict__ D  = dih + (size_t)row * NDIH;
  float* __restrict__ o = out + (size_t)row * NPTS * 3;

  int s0, dir, M, dbase, ddir, obase, odir;
  if (side == 0) { // right: cart[2047..4095], dih[2047..4092], out[2050..4095]
    s0 = SPLIT - 1; dir = +1; M = NPTS - SPLIT - 2;      // 2046 rotations
    dbase = SPLIT - 1; ddir = +1; obase = SPLIT - 1; odir = +1;
  } else {         // left: cart[2049..0], dih[2046..0], out[2049..0]
    s0 = SPLIT + 1; dir = -1; M = SPLIT - 1;             // 2047 rotations
    dbase = SPLIT - 2; ddir = -1; obase = SPLIT + 1; odir = -1;
  }

  // Left half emits its first three (unrotated) points; right's are dropped.
  if (side == 1 && lane < 3) {
    const int g  = obase - lane;
    const int ci = s0 - lane;
    o[3 * g + 0] = rx[ci];
    o[3 * g + 1] = ry[ci];
    o[3 * g + 2] = 0.0f;
  }

  Mat3 P = ident3();                         // global rotation prefix (uniform)
  const int c2 = s0 + dir * 2;
  float posx = rx[c2], posy = ry[c2], posz = 0.0f;

  const int nchunks = (M + 31) / 32;         // 64 for both halves

  // stage chunk 0 dihedrals
  {
    const int m0 = (lane < M) ? lane : (M - 1);
    async_g2l_b32(&D[dbase + ddir * m0], &dstage[wave][0][lane]);
  }

  for (int cch = 0; cch < nchunks; ++cch) {
    const int m = cch * 32 + lane;
    const bool valid = (m < M);

    // stage next chunk's dihedrals; wait for the current chunk's batch
    if (cch + 1 < nchunks) {
      const int nm = m + 32;
      const int cl = (nm < M) ? nm : (M - 1);
      async_g2l_b32(&D[dbase + ddir * cl], &dstage[wave][(cch + 1) % 3][lane]);
      wait_async1();            // current batch retired (in-order completion)
    } else {
      wait_async0();
    }

    // prefetch next chunk's cart operands (gfx1250 global_prefetch_b8)
    {
      const int nm = m + 32;
      if (nm < M) {
        __builtin_prefetch(&rx[s0 + dir * (nm + 2)], 0, 0);
        __builtin_prefetch(&ry[s0 + dir * (nm + 2)], 0, 0);
      }
    }

    Mat3 R = ident3();
    float dvx = 0.0f, dvy = 0.0f;
    const float th_raw = dstage[wave][cch % 3][lane];
    if (valid) {
      const int i1 = s0 + dir * (m + 1);
      const int i2 = s0 + dir * (m + 2);
      const int i3 = s0 + dir * (m + 3);
      const float x1 = rx[i1], y1 = ry[i1];
      const float x2 = rx[i2], y2 = ry[i2];
      const float x3 = rx[i3], y3 = ry[i3];
      float ux = x2 - x1, uy = y2 - y1;
      const float inv = 1.0f / sqrtf(ux * ux + uy * uy);
      ux *= inv; uy *= inv;                  // planar axis: uz == 0
      const float th = th_raw + PI_F;
      float s, c;
      sincosf(th, &s, &c);
      const float omc = 1.0f - c;
      R.m[0][0] = c + omc * ux * ux; R.m[0][1] = omc * ux * uy;     R.m[0][2] =  s * uy;
      R.m[1][0] = omc * ux * uy;     R.m[1][1] = c + omc * uy * uy; R.m[1][2] = -s * ux;
      R.m[2][0] = -s * uy;           R.m[2][1] =  s * ux;           R.m[2][2] =  c;
      dvx = x3 - x2; dvy = y3 - y2;
    }

    // Kogge-Stone inclusive scan: L_i = R_cstart @ ... @ R_i
    #pragma unroll
    for (int off = 1; off < 32; off <<= 1) {
      Mat3 A = shflup3(R, off);
      Mat3 T = matmul3(A, R);
      if (lane >= off) R = T;
    }

    // per-lane global cumulative rotation (32 independent 3x3 -> VALU)
    Mat3 C = matmul3(P, R);

    float ndx = fmaf(C.m[0][0], dvx, C.m[0][1] * dvy);
    float ndy = fmaf(C.m[1][0], dvx, C.m[1][1] * dvy);
    float ndz = fmaf(C.m[2][0], dvx, C.m[2][1] * dvy);

    #pragma unroll
    for (int off = 1; off < 32; off <<= 1) {
      float tx = wup(ndx, off);
      float ty = wup(ndy, off);
      float tz = wup(ndz, off);
      if (lane >= off) { ndx += tx; ndy += ty; ndz += tz; }
    }
    const float px = posx + ndx;
    const float py = posy + ndy;
    const float pz = posz + ndz;

    if (valid) {
      const int t = m + 3;
      const int g = obase + odir * t;
      o[3 * g + 0] = px;
      o[3 * g + 1] = py;
      o[3 * g + 2] = pz;
    }

    // carries: positions from lane 31; rotation prefix via WMMA P @ L31
    posx = wbcast(px, 31);
    posy = wbcast(py, 31);
    posz = wbcast(pz, 31);
    Mat3 L31 = bcast3(R, 31);
    P = carry_update(P, L31, lane);
  }
}

// ---------------------------------------------------------------------------
extern "C" void kernel_launch(void* const* d_in, const int* in_sizes, int n_in,
                              void* d_out, int out_size, void* d_ws, size_t ws_size,
                              hipStream_t stream) {
  const float* distances = (const float*)d_in[0];   // (256, 4095)
  const float* angles    = (const float*)d_in[1];   // (256, 4094)
  const float* dihedrals = (const float*)d_in[2];   // (256, 4093)
  float* out = (float*)d_out;                       // (256, 4096, 3)

  // Workspace layout: lengths[4096] | cartx[256*4096] | carty[256*4096]
  float* w_lengths = (float*)d_ws;
  float* w_cx = w_lengths + NPTS;
  float* w_cy = w_cx + (size_t)BSZ * NPTS;
  (void)in_sizes; (void)n_in; (void)out_size; (void)ws_size;

  k_mean_lengths<<<(NDIST + 255) / 256, 256, 0, stream>>>(distances, w_lengths);
  k_chain<<<BSZ / 8, 256, 0, stream>>>(angles, w_lengths, w_cx, w_cy);
  k_torsion<<<(2 * BSZ) / 8, 256, 0, stream>>>(dihedrals, w_cx, w_cy, out);
}